// GATModel_43542378447395
// MI455X (gfx1250) — compile-verified
//
#include <hip/hip_runtime.h>
#include <math.h>

typedef __attribute__((ext_vector_type(2))) float v2f;
typedef __attribute__((ext_vector_type(8))) float v8f;

#define IN_C  128
#define HID   64
#define HEADS 4
#define F1    (HEADS * HID)   // 256
#define OUT_C 64
#define NEG_SLOPE 0.2f

// ---------------------------------------------------------------------------
// Generic f32 GEMM via V_WMMA_F32_16X16X4_F32.
// One wave computes a 16x16 tile of C = A[M,K] x B[K,Ncol].
// blockDim.x = 128 (4 waves); grid = (ceil(M/16), Ncol/64).
// A-frag (16x4): lanes 0-15 M=0..15 {K+0,K+1}; lanes 16-31 {K+2,K+3}.
// B-frag (4x16): same striping with N across lanes.
// D (16x16 f32): VGPR r -> row r (lanes 0-15) / row r+8 (lanes 16-31).
// ---------------------------------------------------------------------------
__global__ void gat_gemm_wmma_f32(const float* __restrict__ A,
                                  const float* __restrict__ B,
                                  float* __restrict__ C,
                                  int M, int K, int Ncol) {
    const int wid  = threadIdx.x >> 5;
    const int lane = threadIdx.x & 31;
    const int half = lane >> 4;     // 0: K+0/K+1, 1: K+2/K+3
    const int l    = lane & 15;     // row (A) / col (B,D) within tile

    const int rowbase = blockIdx.x * 16;
    const int colbase = blockIdx.y * 64 + wid * 16;

    int arow = rowbase + l;
    if (arow > M - 1) arow = M - 1;             // clamp: keeps EXEC all-ones
    const float* __restrict__ ap = A + (size_t)arow * K;

    v8f acc = {};
    for (int kk = 0; kk < K; kk += 4) {
        const int ka = kk + 2 * half;
        v2f a, b;
        a.x = ap[ka];
        a.y = ap[ka + 1];
        b.x = B[(size_t)ka * Ncol + colbase + l];
        b.y = B[(size_t)(ka + 1) * Ncol + colbase + l];
        acc = __builtin_amdgcn_wmma_f32_16x16x4_f32(
            /*neg_a=*/false, a, /*neg_b=*/false, b,
            /*c_mod=*/(short)0, acc, /*reuse_a=*/false, /*reuse_b=*/false);
    }

#pragma unroll
    for (int r = 0; r < 8; ++r) {
        const int row = rowbase + r + 8 * half;
        if (row < M)
            C[(size_t)row * Ncol + colbase + l] = acc[r];
    }
}

// ---------------------------------------------------------------------------
// Per-(node,head) attention scalars + state init (m = -inf, denom = 0).
// hfeat layout: [(n*H + h) * C]
// ---------------------------------------------------------------------------
__global__ void gat_attn_init(const float* __restrict__ hfeat,
                              const float* __restrict__ a_src,
                              const float* __restrict__ a_dst,
                              float* __restrict__ as, float* __restrict__ ad,
                              float* __restrict__ m, float* __restrict__ denom,
                              int NH, int H, int C) {
    const int i = blockIdx.x * blockDim.x + threadIdx.x;
    if (i >= NH) return;
    const int h = i % H;
    const float* __restrict__ hp = hfeat + (size_t)i * C;
    const float* __restrict__ ws = a_src + (size_t)h * C;
    const float* __restrict__ wd = a_dst + (size_t)h * C;
    float s0 = 0.f, s1 = 0.f;
    for (int c = 0; c < C; ++c) {
        const float v = hp[c];
        s0 = fmaf(v, ws[c], s0);
        s1 = fmaf(v, wd[c], s1);
    }
    as[i] = s0;
    ad[i] = s1;
    m[i] = __int_as_float(0xff800000);   // -inf
    denom[i] = 0.f;
}

__device__ __forceinline__ void atomicMaxFloat(float* addr, float val) {
    // Monotone bit-pattern trick: order-independent, deterministic.
    if (val >= 0.0f)
        atomicMax((int*)addr, __float_as_int(val));
    else
        atomicMin((unsigned int*)addr, __float_as_uint(val));
}

__device__ __forceinline__ void edge_endpoints(const int* __restrict__ ei,
                                               int e, int E, int& s, int& d) {
    if (e < E) { s = ei[e]; d = ei[E + e]; }
    else       { s = d = e - E; }            // appended self-loop
}

// Pass 1: segment max of leaky_relu(alpha_s[src]+alpha_d[dst]) over dst.
__global__ void gat_edge_max(const int* __restrict__ ei, int E, int ET,
                             const float* __restrict__ as,
                             const float* __restrict__ ad,
                             float* __restrict__ m, int H) {
    const int e = blockIdx.x * blockDim.x + threadIdx.x;
    if (e >= ET) return;
    int s, d;
    edge_endpoints(ei, e, E, s, d);
    for (int h = 0; h < H; ++h) {
        float v = as[s * H + h] + ad[d * H + h];
        v = (v > 0.f) ? v : NEG_SLOPE * v;
        atomicMaxFloat(&m[d * H + h], v);
    }
}

// Pass 2: ex = exp(e - m[dst]); cache ex per edge; denom[dst] += ex.
__global__ void gat_edge_exp(const int* __restrict__ ei, int E, int ET,
                             const float* __restrict__ as,
                             const float* __restrict__ ad,
                             const float* __restrict__ m,
                             float* __restrict__ denom,
                             float* __restrict__ ex, int H) {
    const int e = blockIdx.x * blockDim.x + threadIdx.x;
    if (e >= ET) return;
    int s, d;
    edge_endpoints(ei, e, E, s, d);
    for (int h = 0; h < H; ++h) {
        float v = as[s * H + h] + ad[d * H + h];
        v = (v > 0.f) ? v : NEG_SLOPE * v;
        const float ev = __expf(v - m[d * H + h]);
        ex[(size_t)e * H + h] = ev;
        atomicAdd(&denom[d * H + h], ev);
    }
}

// Pass 3: out[dst] += h[src] * (ex / (denom[dst]+eps)).
// One block per edge; blockDim = H*C features.
__global__ void gat_aggregate(const int* __restrict__ ei, int E,
                              const float* __restrict__ hfeat,
                              const float* __restrict__ ex,
                              const float* __restrict__ denom,
                              float* __restrict__ out, int H, int C) {
    const int e = blockIdx.x;
    int s, d;
    edge_endpoints(ei, e, E, s, d);
    const int t = threadIdx.x;          // 0 .. H*C-1
    const int h = t / C;
    const int F = H * C;
    const float alpha = ex[(size_t)e * H + h] / (denom[d * H + h] + 1e-16f);
    atomicAdd(&out[(size_t)d * F + t], hfeat[(size_t)s * F + t] * alpha);
}

__global__ void gat_zero(float* __restrict__ p, long long n) {
    const long long i = (long long)blockIdx.x * blockDim.x + threadIdx.x;
    if (i < n) p[i] = 0.f;
}

// out = elu(out + bias)  (in place; feeds layer 2)
__global__ void gat_bias_elu(float* __restrict__ out, const float* __restrict__ b,
                             long long total, int F) {
    const long long i = (long long)blockIdx.x * blockDim.x + threadIdx.x;
    if (i >= total) return;
    const float v = out[i] + b[i % F];
    out[i] = (v > 0.f) ? v : (__expf(v) - 1.0f);
}

__global__ void gat_add_bias(float* __restrict__ out, const float* __restrict__ b,
                             long long total, int F) {
    const long long i = (long long)blockIdx.x * blockDim.x + threadIdx.x;
    if (i >= total) return;
    out[i] += b[i % F];
}

// ---------------------------------------------------------------------------
extern "C" void kernel_launch(void* const* d_in, const int* in_sizes, int n_in,
                              void* d_out, int out_size, void* d_ws, size_t ws_size,
                              hipStream_t stream) {
    const float* x    = (const float*)d_in[0];
    const int*   ei   = (const int*)  d_in[1];
    const float* W1   = (const float*)d_in[2];
    const float* a1s  = (const float*)d_in[3];
    const float* a1d  = (const float*)d_in[4];
    const float* b1   = (const float*)d_in[5];
    const float* W2   = (const float*)d_in[6];
    const float* a2s  = (const float*)d_in[7];
    const float* a2d  = (const float*)d_in[8];
    const float* b2   = (const float*)d_in[9];

    const int Nn = in_sizes[0] / IN_C;   // 50000
    const int E  = in_sizes[1] / 2;      // 800000
    const int ET = E + Nn;               // + self loops

    // ---- workspace carve (floats) ----
    float* ws   = (float*)d_ws;
    float* h1   = ws;                          // Nn * F1
    float* out1 = h1   + (size_t)Nn * F1;      // Nn * F1 (accum -> act1 in place)
    float* as1  = out1 + (size_t)Nn * F1;      // Nn * HEADS
    float* ad1  = as1  + (size_t)Nn * HEADS;
    float* m1   = ad1  + (size_t)Nn * HEADS;
    float* dn1  = m1   + (size_t)Nn * HEADS;
    float* ex1  = dn1  + (size_t)Nn * HEADS;   // ET * HEADS
    float* h2   = ex1  + (size_t)ET * HEADS;   // Nn * OUT_C
    float* as2  = h2   + (size_t)Nn * OUT_C;   // Nn
    float* ad2  = as2  + (size_t)Nn;
    float* m2   = ad2  + (size_t)Nn;
    float* dn2  = m2   + (size_t)Nn;
    float* ex2  = dn2  + (size_t)Nn;           // ET

    const int rowTiles = (Nn + 15) / 16;
    float* dout = (float*)d_out;

    // ================= Layer 1 =================
    {
        const long long tot = (long long)Nn * F1;
        gat_zero<<<dim3((unsigned)((tot + 255) / 256)), 256, 0, stream>>>(out1, tot);
    }
    gat_gemm_wmma_f32<<<dim3(rowTiles, F1 / 64), 128, 0, stream>>>(x, W1, h1, Nn, IN_C, F1);
    gat_attn_init<<<(Nn * HEADS + 255) / 256, 256, 0, stream>>>(h1, a1s, a1d,
                                                                as1, ad1, m1, dn1,
                                                                Nn * HEADS, HEADS, HID);
    gat_edge_max<<<(ET + 255) / 256, 256, 0, stream>>>(ei, E, ET, as1, ad1, m1, HEADS);
    gat_edge_exp<<<(ET + 255) / 256, 256, 0, stream>>>(ei, E, ET, as1, ad1, m1, dn1, ex1, HEADS);
    gat_aggregate<<<ET, F1, 0, stream>>>(ei, E, h1, ex1, dn1, out1, HEADS, HID);
    {
        const long long tot = (long long)Nn * F1;
        gat_bias_elu<<<(unsigned)((tot + 255) / 256), 256, 0, stream>>>(out1, b1, tot, F1);
    }

    // ================= Layer 2 =================
    gat_gemm_wmma_f32<<<dim3(rowTiles, OUT_C / 64), 128, 0, stream>>>(out1, W2, h2, Nn, F1, OUT_C);
    gat_attn_init<<<(Nn + 255) / 256, 256, 0, stream>>>(h2, a2s, a2d,
                                                        as2, ad2, m2, dn2,
                                                        Nn, 1, OUT_C);
    {
        const long long tot = (long long)Nn * OUT_C;
        gat_zero<<<(unsigned)((tot + 255) / 256), 256, 0, stream>>>(dout, tot);
    }
    gat_edge_max<<<(ET + 255) / 256, 256, 0, stream>>>(ei, E, ET, as2, ad2, m2, 1);
    gat_edge_exp<<<(ET + 255) / 256, 256, 0, stream>>>(ei, E, ET, as2, ad2, m2, dn2, ex2, 1);
    gat_aggregate<<<ET, OUT_C, 0, stream>>>(ei, E, h2, ex2, dn2, dout, 1, OUT_C);
    {
        const long long tot = (long long)Nn * OUT_C;
        gat_add_bias<<<(unsigned)((tot + 255) / 256), 256, 0, stream>>>(dout, b2, tot, OUT_C);
    }
}